// Qwen3NextDecoderLayer_25245817766044
// MI455X (gfx1250) — compile-verified
//
#include <hip/hip_runtime.h>
#include <math.h>

typedef float v2f __attribute__((ext_vector_type(2)));
typedef float v4f __attribute__((ext_vector_type(4)));
typedef float v8f __attribute__((ext_vector_type(8)));

#define T_TOK 4096
#define DMODEL 2048
#define LSEQ 2048
#define NB 2
#define QKVZ_N 12288
#define CONV_N 8192
#define HKH 16
#define HVH 32
#define IFF 5120
#define EPSF 1e-6f

__device__ __forceinline__ float siluf(float x) { return x / (1.f + expf(-x)); }

// ---------------------------------------------------------------------------
// CDNA5 async global->LDS copy support (ASYNCcnt-tracked). Falls back to the
// synchronous staging if the builtin is absent.
// Probe round 2 revealed the builtin signature: 4 args, first param is
// 'int __vector(4) __device__ *' (int4 addrspace(1)*), so we cast to the exact
// GCC-vector pointer types in the right address spaces.
// ---------------------------------------------------------------------------
#if defined(__has_builtin)
#if __has_builtin(__builtin_amdgcn_global_load_async_to_lds_b128)
#define HAVE_ASYNC_LDS 1
#endif
#endif

#ifdef HAVE_ASYNC_LDS
typedef int vsi4 __attribute__((vector_size(16)));  // GCC-style, matches builtin
#define GLOBAL_AS __attribute__((address_space(1)))
#define LDS_AS __attribute__((address_space(3)))
__device__ __forceinline__ void async_cp16(void* lds, const void* g) {
  // copies 16 bytes global -> LDS, per lane, tracked by ASYNCcnt
  __builtin_amdgcn_global_load_async_to_lds_b128(
      (GLOBAL_AS vsi4*)g, (LDS_AS vsi4*)lds, 0, 0);
}
__device__ __forceinline__ void wait_async0() {
#if __has_builtin(__builtin_amdgcn_s_wait_asynccnt)
  __builtin_amdgcn_s_wait_asynccnt(0);
#else
  asm volatile("s_wait_asynccnt 0x0" ::: "memory");
#endif
}
#endif

// ---------------------------------------------------------------------------
// Generic fp32 WMMA GEMM: C[M,N] = A[M,K] @ B[K,N] (+ Res[M,N] if Res != null)
// Row-major, lda=K, ldb=ldc=N. M % 128 == 0, K % 16 == 0, N guarded.
// Tile 128x128x16; 256 threads = 8 waves; each wave owns 32(M) x 64(N)
// via 2x4 accumulators of V_WMMA_F32_16X16X4_F32.
// Double-buffered LDS; tiles staged with GLOBAL_LOAD_ASYNC_TO_LDS_B128 so the
// next tile's HBM fetch overlaps the current tile's WMMA stream.
// ---------------------------------------------------------------------------
__global__ __launch_bounds__(256) void gemm_f32_wmma(
    const float* __restrict__ A, const float* __restrict__ Bm,
    const float* __restrict__ Res, float* __restrict__ C,
    int M, int N, int K) {
  __shared__ float As[2][128 * 16];
  __shared__ float Bs[2][16 * 128];
  const int tid = threadIdx.x;
  const int lane = tid & 31;
  const int l16 = lane & 15;
  const int lhalf = lane >> 4;
  const int wave = tid >> 5;
  const int wmBase = (wave & 3) * 32;   // 4 waves along M
  const int wnBase = (wave >> 2) * 64;  // 2 waves along N
  const int mBase = blockIdx.y * 128;
  const int nBase = blockIdx.x * 128;
  const bool fullN = (nBase + 128 <= N);  // uniform per block

  v8f acc[2][4];
#pragma unroll
  for (int mt = 0; mt < 2; ++mt)
#pragma unroll
    for (int nt = 0; nt < 4; ++nt)
#pragma unroll
      for (int r = 0; r < 8; ++r) acc[mt][nt][r] = 0.f;

  // ---- tile staging (async when available, sync fallback) ----
  auto stage = [&](int k0, int buf) {
    // A tile (128x16): always fully in-bounds (M%128==0, K%16==0)
#ifdef HAVE_ASYNC_LDS
#pragma unroll
    for (int i = 0; i < 2; ++i) {
      int lin = tid + i * 256;          // 0..511
      int row = lin >> 2;               // 0..127
      int kq = (lin & 3) << 2;          // 0,4,8,12
      async_cp16(&As[buf][row * 16 + kq],
                 A + (size_t)(mBase + row) * K + k0 + kq);
    }
#else
#pragma unroll
    for (int i = 0; i < 2; ++i) {
      int lin = tid + i * 256;
      int row = lin >> 2;
      int kq = (lin & 3) << 2;
      v4f va = *reinterpret_cast<const v4f*>(
          A + (size_t)(mBase + row) * K + k0 + kq);
      *reinterpret_cast<v4f*>(&As[buf][row * 16 + kq]) = va;
    }
#endif
    // B tile (16x128)
    if (fullN) {
#ifdef HAVE_ASYNC_LDS
#pragma unroll
      for (int i = 0; i < 2; ++i) {
        int lin = tid + i * 256;
        int row = lin >> 5;             // 0..15
        int nq = (lin & 31) << 2;       // 0..124
        async_cp16(&Bs[buf][row * 128 + nq],
                   Bm + (size_t)(k0 + row) * N + nBase + nq);
      }
#else
#pragma unroll
      for (int i = 0; i < 2; ++i) {
        int lin = tid + i * 256;
        int row = lin >> 5;
        int nq = (lin & 31) << 2;
        v4f vb = *reinterpret_cast<const v4f*>(
            Bm + (size_t)(k0 + row) * N + nBase + nq);
        *reinterpret_cast<v4f*>(&Bs[buf][row * 128 + nq]) = vb;
      }
#endif
    } else {
      // N-guarded sync path (only the N=64 W_ba GEMM hits this)
#pragma unroll
      for (int i = 0; i < 2; ++i) {
        int lin = tid + i * 256;
        int row = lin >> 5;
        int nq = (lin & 31) << 2;
        int gn = nBase + nq;
        const float* bp = Bm + (size_t)(k0 + row) * N + gn;
        v4f vb;
        vb[0] = (gn + 0 < N) ? bp[0] : 0.f;
        vb[1] = (gn + 1 < N) ? bp[1] : 0.f;
        vb[2] = (gn + 2 < N) ? bp[2] : 0.f;
        vb[3] = (gn + 3 < N) ? bp[3] : 0.f;
        *reinterpret_cast<v4f*>(&Bs[buf][row * 128 + nq]) = vb;
      }
    }
  };

  // ---- software pipeline: stage(k+1) overlaps compute(k) ----
  int buf = 0;
  stage(0, 0);
#ifdef HAVE_ASYNC_LDS
  wait_async0();
#endif
  __syncthreads();

  for (int k0 = 0; k0 < K; k0 += 16) {
    if (k0 + 16 < K) {
      stage(k0 + 16, buf ^ 1);
      // pull the tile after next toward L2 while we compute
      __builtin_prefetch(A + (size_t)(mBase + (tid >> 1)) * K + k0 + 32, 0, 0);
      __builtin_prefetch(Bm + (size_t)(k0 + 32 + (tid >> 5)) * N + nBase + (tid & 31) * 4, 0, 0);
    }

    const float* Asb = As[buf];
    const float* Bsb = Bs[buf];
#pragma unroll
    for (int kk = 0; kk < 4; ++kk) {
      v2f aF[2];
      v2f bF[4];
#pragma unroll
      for (int mt = 0; mt < 2; ++mt) {
        const float* p = &Asb[(wmBase + mt * 16 + l16) * 16 + kk * 4 + 2 * lhalf];
        aF[mt][0] = p[0];
        aF[mt][1] = p[1];
      }
#pragma unroll
      for (int nt = 0; nt < 4; ++nt) {
        int col = wnBase + nt * 16 + l16;
        bF[nt][0] = Bsb[(kk * 4 + 2 * lhalf + 0) * 128 + col];
        bF[nt][1] = Bsb[(kk * 4 + 2 * lhalf + 1) * 128 + col];
      }
#pragma unroll
      for (int mt = 0; mt < 2; ++mt)
#pragma unroll
        for (int nt = 0; nt < 4; ++nt)
          acc[mt][nt] = __builtin_amdgcn_wmma_f32_16x16x4_f32(
              false, aF[mt], false, bF[nt], (short)0, acc[mt][nt], false, false);
    }

#ifdef HAVE_ASYNC_LDS
    wait_async0();  // next buffer's async copies have landed in LDS
#endif
    __syncthreads();  // all waves done reading current buffer
    buf ^= 1;
  }

  // ---- epilogue: C layout VGPR r -> M = r + 8*lhalf, N = l16 ----
#pragma unroll
  for (int mt = 0; mt < 2; ++mt) {
#pragma unroll
    for (int nt = 0; nt < 4; ++nt) {
      int gcol = nBase + wnBase + nt * 16 + l16;
      if (gcol < N) {
#pragma unroll
        for (int r = 0; r < 8; ++r) {
          size_t grow = (size_t)(mBase + wmBase + mt * 16 + r + 8 * lhalf);
          size_t o = grow * (size_t)N + gcol;
          float v = acc[mt][nt][r];
          if (Res) v += Res[o];
          C[o] = v;
        }
      }
    }
  }
}

// ---------------------------------------------------------------------------
// RMSNorm over last dim D, one token per block.
// ---------------------------------------------------------------------------
__global__ __launch_bounds__(256) void rmsnorm_kernel(
    const float* __restrict__ x, const float* __restrict__ w,
    float* __restrict__ y, int D) {
  const size_t t = blockIdx.x;
  const int tid = threadIdx.x;
  const float* xr = x + t * D;
  float ss = 0.f;
  for (int i = tid; i < D; i += 256) { float v = xr[i]; ss += v * v; }
  __shared__ float red[256];
  red[tid] = ss;
  __syncthreads();
  for (int s = 128; s > 0; s >>= 1) {
    if (tid < s) red[tid] += red[tid + s];
    __syncthreads();
  }
  float inv = rsqrtf(red[0] / (float)D + EPSF);
  for (int i = tid; i < D; i += 256) y[t * D + i] = xr[i] * inv * w[i];
}

// ---------------------------------------------------------------------------
// Causal depthwise conv (K=4) + SiLU, reading q/k/v directly from qkvz layout.
// ---------------------------------------------------------------------------
__global__ __launch_bounds__(256) void conv_silu_kernel(
    const float* __restrict__ qkvz, const float* __restrict__ cw,
    float* __restrict__ conv) {
  const int t = blockIdx.x;
  const int c = blockIdx.y * 256 + threadIdx.x;
  const int b = t / LSEQ, l = t % LSEQ;
  int col;
  if (c < 2048) {
    int hk = c >> 7, d = c & 127;
    col = hk * 768 + d;                               // q
  } else if (c < 4096) {
    int cc = c - 2048; int hk = cc >> 7, d = cc & 127;
    col = hk * 768 + 128 + d;                         // k
  } else {
    int cc = c - 4096; int hv = cc >> 7, d = cc & 127;
    col = (hv >> 1) * 768 + 256 + (hv & 1) * 128 + d; // v
  }
  float acc = 0.f;
#pragma unroll
  for (int i = 0; i < 4; ++i) {
    int ll = l - 3 + i;
    if (ll >= 0)
      acc = fmaf(qkvz[(size_t)(b * LSEQ + ll) * QKVZ_N + col], cw[c * 4 + i], acc);
  }
  conv[(size_t)t * CONV_N + c] = siluf(acc);
}

// ---------------------------------------------------------------------------
// Gather z out of qkvz so the qkvz buffer can be recycled.
// ---------------------------------------------------------------------------
__global__ __launch_bounds__(256) void zgather_kernel(
    const float* __restrict__ qkvz, float* __restrict__ zbuf) {
  size_t idx = (size_t)blockIdx.x * 256 + threadIdx.x;  // t*4096 + hv*128 + d
  int t = (int)(idx >> 12);
  int c = (int)(idx & 4095);
  int hv = c >> 7, d = c & 127;
  zbuf[idx] = qkvz[(size_t)t * QKVZ_N + (hv >> 1) * 768 + 512 + (hv & 1) * 128 + d];
}

// ---------------------------------------------------------------------------
// l2norm(q)*DK^-0.5 and l2norm(k), repeated HK->HV; plus g = -exp(A)*softplus,
// beta = sigmoid(b). One token per block, one wave32 per HK head.
// ---------------------------------------------------------------------------
__global__ __launch_bounds__(512) void qknorm_gb_kernel(
    const float* __restrict__ conv, const float* __restrict__ ba,
    const float* __restrict__ alog, const float* __restrict__ dtb,
    float* __restrict__ qn, float* __restrict__ kn,
    float* __restrict__ gb, float* __restrict__ bb) {
  const size_t t = blockIdx.x;
  const int tid = threadIdx.x;
  const int hk = tid >> 5, lane = tid & 31;

  // q head
  v4f qv = *reinterpret_cast<const v4f*>(conv + t * CONV_N + hk * 128 + lane * 4);
  float ss = qv[0] * qv[0] + qv[1] * qv[1] + qv[2] * qv[2] + qv[3] * qv[3];
  for (int m = 16; m >= 1; m >>= 1) ss += __shfl_xor(ss, m, 32);
  float qs = rsqrtf(ss + EPSF) * 0.08838834764831845f;  // DK^-0.5
  size_t o0 = (t * 32 + 2 * hk) * 128 + lane * 4;
  v4f qo = {qv[0] * qs, qv[1] * qs, qv[2] * qs, qv[3] * qs};
  *reinterpret_cast<v4f*>(qn + o0) = qo;
  *reinterpret_cast<v4f*>(qn + o0 + 128) = qo;   // repeat to hv = 2*hk+1

  // k head
  v4f kv = *reinterpret_cast<const v4f*>(conv + t * CONV_N + 2048 + hk * 128 + lane * 4);
  float ks = kv[0] * kv[0] + kv[1] * kv[1] + kv[2] * kv[2] + kv[3] * kv[3];
  for (int m = 16; m >= 1; m >>= 1) ks += __shfl_xor(ks, m, 32);
  float kc = rsqrtf(ks + EPSF);
  v4f ko = {kv[0] * kc, kv[1] * kc, kv[2] * kc, kv[3] * kc};
  *reinterpret_cast<v4f*>(kn + o0) = ko;
  *reinterpret_cast<v4f*>(kn + o0 + 128) = ko;

  if (tid < 32) {
    int hv = tid;
    float bv = ba[t * 64 + (hv >> 1) * 4 + (hv & 1)];
    float av = ba[t * 64 + (hv >> 1) * 4 + 2 + (hv & 1)];
    float x = av + dtb[hv];
    float sp = (x > 20.f) ? x : log1pf(expf(x));
    gb[t * 32 + hv] = -expf(alog[hv]) * sp;
    bb[t * 32 + hv] = 1.f / (1.f + expf(-bv));
  }
}

// ---------------------------------------------------------------------------
// Gated delta-rule scan. One block per (b, hv); 128x128 fp32 state held in
// registers (64 floats/thread, 256 threads); k/q/v broadcast through LDS.
// ---------------------------------------------------------------------------
__global__ __launch_bounds__(256) void delta_scan_kernel(
    const float* __restrict__ qn, const float* __restrict__ kn,
    const float* __restrict__ conv, const float* __restrict__ gb,
    const float* __restrict__ bb, float* __restrict__ core) {
  const int b = blockIdx.x >> 5;
  const int hv = blockIdx.x & 31;
  const int tid = threadIdx.x;
  const int col = tid & 127;
  const int half = tid >> 7;
  const int dk0 = half * 64;
  __shared__ float sq[128], sk[128], sv[128];
  __shared__ float part[2][128];
  __shared__ float sgb[2];
  float S[64];
#pragma unroll
  for (int j = 0; j < 64; ++j) S[j] = 0.f;

  for (int l = 0; l < LSEQ; ++l) {
    const size_t t = (size_t)b * LSEQ + l;
    const size_t qk = (t * 32 + hv) * 128;
    if (half == 0) {
      sq[col] = qn[qk + col];
      sv[col] = conv[t * CONV_N + 4096 + hv * 128 + col];
    } else {
      sk[col] = kn[qk + col];
    }
    if (tid == 0) {
      sgb[0] = expf(gb[t * 32 + hv]);
      sgb[1] = bb[t * 32 + hv];
    }
    __syncthreads();  // B1
    const float eg = sgb[0], bt = sgb[1];
    float pkv = 0.f;
#pragma unroll
    for (int j = 0; j < 64; ++j) {
      S[j] *= eg;
      pkv = fmaf(sk[dk0 + j], S[j], pkv);
    }
    part[half][col] = pkv;
    __syncthreads();  // B2
    const float delta = (sv[col] - (part[0][col] + part[1][col])) * bt;
    __syncthreads();  // B3: all reads of part done before overwrite
    float pout = 0.f;
#pragma unroll
    for (int j = 0; j < 64; ++j) {
      S[j] = fmaf(sk[dk0 + j], delta, S[j]);
      pout = fmaf(sq[dk0 + j], S[j], pout);
    }
    part[half][col] = pout;
    __syncthreads();  // B4
    if (half == 0) core[qk + col] = part[0][col] + part[1][col];
    __syncthreads();  // B5: protect shared buffers before next iteration
  }
}

// ---------------------------------------------------------------------------
// core = rmsnorm(core, gw) * silu(z), in place. One (t, hv) per block.
// ---------------------------------------------------------------------------
__global__ __launch_bounds__(128) void gatednorm_kernel(
    float* __restrict__ core, const float* __restrict__ zbuf,
    const float* __restrict__ gw) {
  const size_t th = blockIdx.x;  // t*32 + hv
  const int d = threadIdx.x;
  __shared__ float red[128];
  const size_t base = th * 128;
  float v = core[base + d];
  red[d] = v * v;
  __syncthreads();
  for (int s = 64; s > 0; s >>= 1) {
    if (d < s) red[d] += red[d + s];
    __syncthreads();
  }
  float inv = rsqrtf(red[0] * (1.f / 128.f) + EPSF);
  float z = zbuf[base + d];
  core[base + d] = v * inv * gw[d] * siluf(z);
}

__global__ __launch_bounds__(256) void silu_mul_kernel(
    float* __restrict__ g, const float* __restrict__ u, size_t n) {
  size_t i = (size_t)blockIdx.x * 256 + threadIdx.x;
  if (i < n) { float x = g[i]; g[i] = siluf(x) * u[i]; }
}

static void launch_gemm(const float* A, const float* B, const float* Res,
                        float* C, int M, int N, int K, hipStream_t s) {
  dim3 grid((N + 127) / 128, M / 128);
  gemm_f32_wmma<<<grid, 256, 0, s>>>(A, B, Res, C, M, N, K);
}

extern "C" void kernel_launch(void* const* d_in, const int* in_sizes, int n_in,
                              void* d_out, int out_size, void* d_ws, size_t ws_size,
                              hipStream_t stream) {
  const float* hidden = (const float*)d_in[0];
  const float* ln1 = (const float*)d_in[1];
  const float* Wqkvz = (const float*)d_in[2];
  const float* Wba = (const float*)d_in[3];
  const float* convw = (const float*)d_in[4];
  const float* dtb = (const float*)d_in[5];
  const float* alog = (const float*)d_in[6];
  const float* gnw = (const float*)d_in[7];
  const float* Wout = (const float*)d_in[8];
  const float* ln2 = (const float*)d_in[9];
  const float* Wg = (const float*)d_in[10];
  const float* Wu = (const float*)d_in[11];
  const float* Wd = (const float*)d_in[12];
  float* out = (float*)d_out;
  float* ws = (float*)d_ws;

  // Workspace layout (floats). Peak = 109,314,048 floats ~= 437 MB.
  float* h1   = ws;                    // 8,388,608  (T x D)
  float* qkvz = ws + 8388608ull;       // 50,331,648 (T x 12288)
  float* ba   = ws + 58720256ull;      // 262,144    (T x 64)
  float* conv = ws + 58982400ull;      // 33,554,432 (T x 8192)
  float* zbuf = ws + 92536832ull;      // 16,777,216 (T x 4096)
  // After conv+zgather, the qkvz region is recycled:
  float* qn   = ws + 8388608ull;       // 16,777,216 (T x 32 x 128)
  float* kn   = ws + 25165824ull;      // 16,777,216
  float* gbuf = ws + 41943040ull;      // 131,072
  float* bbuf = ws + 42074112ull;      // 131,072
  float* core = ws + 42205184ull;      // 16,777,216 (ends exactly at conv base)
  // After the scan, conv region is recycled:
  float* hbuf = ws + 58982400ull;      // 8,388,608 (residual stream h)
  float* h2   = ws;                    // reuse h1
  // After attention-out GEMM, qn/kn/core regions are recycled:
  float* gate = ws + 8388608ull;       // 20,971,520 (T x 5120)
  float* up   = ws + 29360128ull;      // 20,971,520

  // 1) h1 = rmsnorm(hidden, ln1)
  rmsnorm_kernel<<<T_TOK, 256, 0, stream>>>(hidden, ln1, h1, DMODEL);
  // 2) qkvz = h1 @ W_qkvz ; ba = h1 @ W_ba
  launch_gemm(h1, Wqkvz, nullptr, qkvz, T_TOK, QKVZ_N, DMODEL, stream);
  launch_gemm(h1, Wba, nullptr, ba, T_TOK, 64, DMODEL, stream);
  // 3) causal conv + silu; stash z
  conv_silu_kernel<<<dim3(T_TOK, CONV_N / 256), 256, 0, stream>>>(qkvz, convw, conv);
  zgather_kernel<<<(T_TOK * 4096) / 256, 256, 0, stream>>>(qkvz, zbuf);
  // 4) q/k l2norm (+repeat, +scale), g/beta
  qknorm_gb_kernel<<<T_TOK, 512, 0, stream>>>(conv, ba, alog, dtb, qn, kn, gbuf, bbuf);
  // 5) gated delta-rule scan
  delta_scan_kernel<<<NB * HVH, 256, 0, stream>>>(qn, kn, conv, gbuf, bbuf, core);
  // 6) gated rmsnorm * silu(z), in place
  gatednorm_kernel<<<T_TOK * HVH, 128, 0, stream>>>(core, zbuf, gnw);
  // 7) h = hidden + core @ W_out
  launch_gemm(core, Wout, hidden, hbuf, T_TOK, DMODEL, 4096, stream);
  // 8) h2 = rmsnorm(h, ln2)
  rmsnorm_kernel<<<T_TOK, 256, 0, stream>>>(hbuf, ln2, h2, DMODEL);
  // 9) gate/up GEMMs, SwiGLU, down-proj + residual -> out
  launch_gemm(h2, Wg, nullptr, gate, T_TOK, IFF, DMODEL, stream);
  launch_gemm(h2, Wu, nullptr, up, T_TOK, IFF, DMODEL, stream);
  silu_mul_kernel<<<(T_TOK * IFF) / 256, 256, 0, stream>>>(gate, up, (size_t)T_TOK * IFF);
  launch_gemm(gate, Wd, hbuf, out, T_TOK, DMODEL, IFF, stream);
}